// EpisA_24111946399758
// MI455X (gfx1250) — compile-verified
//
#include <hip/hip_runtime.h>

typedef float v2f __attribute__((ext_vector_type(2)));
typedef float v8f __attribute__((ext_vector_type(8)));

#define NN 128
#define HH 2
#define LL 1024
#define CC (HH * LL)   // 2048 columns of the alpha matrix

// ---------------------------------------------------------------------------
// Kernel 1: per-(n,h)-row scans. One wave32 per row (256 rows total).
//   signal = relu(x)                       -> d_out[signal slot]
//   Ss     = 1 - inclusive_cumsum(signal)  -> d_out[predSignal slot] (temp)
//   Is[i]  = r*Is[i-1] + signal[i]  (r = 1 - 1/tau), via affine-scan
//   alpha  = 1 - exp(-R0dTau * Is)         -> workspace
// ---------------------------------------------------------------------------
__global__ void __launch_bounds__(256)
scan_kernel(const float* __restrict__ x,
            const float* __restrict__ taus,
            const float* __restrict__ r0d,
            float* __restrict__ Ss_out,
            float* __restrict__ signal_out,
            float* __restrict__ alpha_ws)
{
    int wave = (int)((blockIdx.x * blockDim.x + threadIdx.x) >> 5);
    int lane = (int)(threadIdx.x & 31);
    if (wave >= NN * HH) return;

    float tau = taus[wave];
    float r   = 1.0f - 1.0f / tau;
    float rd  = r0d[wave];

    const float* xr = x + (size_t)wave * LL;
    float carry_sum = 0.0f;
    float carry_is  = 0.0f;

    for (int chunk = 0; chunk < LL / 32; ++chunk) {
        int i = chunk * 32 + lane;
        float s = xr[i];
        s = s > 0.0f ? s : 0.0f;
        signal_out[(size_t)wave * LL + i] = s;

        // --- inclusive sum scan (Kogge-Stone over the wave) ---
        float ssum = s;
        #pragma unroll
        for (int off = 1; off < 32; off <<= 1) {
            float t = __shfl_up(ssum, off, 32);
            if (lane >= off) ssum += t;
        }
        float cum = carry_sum + ssum;
        Ss_out[(size_t)wave * LL + i] = 1.0f - cum;
        carry_sum += __shfl(ssum, 31, 32);

        // --- affine recurrence scan: state transform x -> A*x + B ---
        // element transform: (A,B) = (r, s); compose prev-then-self:
        //   B = A*Bprev + B ; A = A*Aprev
        float A = r, B = s;
        #pragma unroll
        for (int off = 1; off < 32; off <<= 1) {
            float Ap = __shfl_up(A, off, 32);
            float Bp = __shfl_up(B, off, 32);
            if (lane >= off) { B = A * Bp + B; A = A * Ap; }
        }
        float Is = A * carry_is + B;
        carry_is = __shfl(Is, 31, 32);

        alpha_ws[(size_t)wave * LL + i] = 1.0f - __expf(-rd * Is);
    }
}

// ---------------------------------------------------------------------------
// Kernel 2: third output, tempAmat.T = Amat + I  (128x128).
// Also serves as the precomputed A-operand matrix for the WMMA kernel.
// ---------------------------------------------------------------------------
__global__ void __launch_bounds__(256)
tmat_kernel(const float* __restrict__ Amat, float* __restrict__ out)
{
    int idx = (int)(blockIdx.x * blockDim.x + threadIdx.x);
    if (idx >= NN * NN) return;
    int i = idx >> 7;
    int j = idx & 127;
    out[idx] = Amat[idx] + (i == j ? 1.0f : 0.0f);
}

// ---------------------------------------------------------------------------
// Kernel 3: Alpha = T @ alpha  via V_WMMA_F32_16X16X4_F32 (T = Amat + I,
// read precomputed from the tempAmat.T output slot), then
// predSignal = Alpha * Ss written in place over the Ss temp.
// One wave per 16x16 output tile; 8x128 tiles; K=128 in 32 WMMA steps.
//
// A-operand (16x4 f32): lanes 0-15 supply M=lane, K={0,1}; lanes 16-31
// supply M=lane-16, K={2,3} (2 VGPRs per lane) -> one aligned float2 load.
// B-operand (4x16) mirrors: lane half selects K half, VGPR steps K, N=lane%16.
// C/D (16x16 f32): VGPR rr holds M = rr + 8*(lane>=16), N = lane%16.
// ---------------------------------------------------------------------------
__global__ void __launch_bounds__(128)
mm_kernel(const float* __restrict__ Tmat,
          const float* __restrict__ alpha_ws,
          float* __restrict__ pred /* holds Ss on entry, predSignal on exit */)
{
    int waveInBlk = (int)(threadIdx.x >> 5);
    int lane      = (int)(threadIdx.x & 31);
    int gwave     = (int)blockIdx.x * 4 + waveInBlk;   // 0..1023, exact fit

    int mt = gwave >> 7;     // tile row   0..7   (k0 = mt*16)
    int nt = gwave & 127;    // tile col   0..127 (c0 = nt*16)
    int k0 = mt * 16;
    int c0 = nt * 16;
    int half = lane >> 4;    // which K-pair this lane supplies
    int lq   = lane & 15;

    v8f acc = {0.f, 0.f, 0.f, 0.f, 0.f, 0.f, 0.f, 0.f};

    int arow = k0 + lq;                           // A-matrix row for this lane
    const float* Trow   = Tmat + (size_t)arow * NN;
    const float* colptr = alpha_ws + c0 + lq;

    for (int m0 = 0; m0 < NN; m0 += 4) {
        int ma = m0 + 2 * half;                   // even -> 8B-aligned float2
        v2f a = *(const v2f*)(Trow + ma);
        v2f b;
        b.x = colptr[(size_t)(ma)     * CC];
        b.y = colptr[(size_t)(ma + 1) * CC];
        acc = __builtin_amdgcn_wmma_f32_16x16x4_f32(
                  /*neg_a=*/false, a, /*neg_b=*/false, b,
                  /*c_mod=*/(short)0, acc, /*reuse_a=*/false, /*reuse_b=*/false);
    }

    #pragma unroll
    for (int rr = 0; rr < 8; ++rr) {
        int k = k0 + rr + 8 * half;
        size_t idx = (size_t)k * CC + c0 + lq;
        float ssv = pred[idx];      // Ss value staged there by scan_kernel
        pred[idx] = acc[rr] * ssv;  // predSignal
    }
}

// ---------------------------------------------------------------------------
// Launch
//   d_in[0] = x       (128*2*1024 f32)
//   d_in[1] = Amat    (128*128 f32)
//   d_in[2] = taus    (128*2 f32)
//   d_in[3] = R0dTaus (128*2 f32)
//   d_out   = [predSignal (262144) | signal (262144) | tempAmat.T (16384)] f32
//   d_ws    = alpha scratch (128*2048 f32 = 1 MB)
// ---------------------------------------------------------------------------
extern "C" void kernel_launch(void* const* d_in, const int* in_sizes, int n_in,
                              void* d_out, int out_size, void* d_ws, size_t ws_size,
                              hipStream_t stream)
{
    const float* x    = (const float*)d_in[0];
    const float* Amat = (const float*)d_in[1];
    const float* taus = (const float*)d_in[2];
    const float* r0d  = (const float*)d_in[3];

    float* out      = (float*)d_out;
    float* Ss_slot  = out;                         // predSignal slot (temp Ss)
    float* sig_slot = out + (size_t)NN * HH * LL;  // signal output
    float* tmat     = out + (size_t)2 * NN * HH * LL;
    float* alpha_ws = (float*)d_ws;                // 128*2048 f32

    // 256 rows, one wave each -> 32 blocks of 256 threads (8 waves)
    scan_kernel<<<32, 256, 0, stream>>>(x, taus, r0d, Ss_slot, sig_slot, alpha_ws);

    // 16384 elements: T = Amat + I (output #3 and WMMA A-operand)
    tmat_kernel<<<64, 256, 0, stream>>>(Amat, tmat);

    // 1024 tiles, 4 waves per 128-thread block -> 256 blocks
    mm_kernel<<<256, 128, 0, stream>>>(tmat, alpha_ws, Ss_slot);
}